// GSTA_41394894799544
// MI455X (gfx1250) — compile-verified
//
#include <hip/hip_runtime.h>
#include <stdint.h>

#define LSE_BETA 10.0f

// ---------- order-preserving float <-> u32 key (for exact atomic max) ----------
__device__ __forceinline__ unsigned fkey(float f) {
  unsigned b = __float_as_uint(f);
  return (b & 0x80000000u) ? ~b : (b | 0x80000000u);
}
__device__ __forceinline__ float fkey_inv(unsigned k) {
  unsigned b = (k & 0x80000000u) ? (k & 0x7fffffffu) : ~k;
  return __uint_as_float(b);
}

// ---------- branchless locate on an ascending 8-point axis ----------
// idx = clip(count(ax < x) - 1, 0, 6); t = (x - ax[idx]) / (ax[idx+1] - ax[idx])
__device__ __forceinline__ void locate8(float4 v0, float4 v1, float x,
                                        int& idx, float& tf) {
  float a0 = v0.x, a1 = v0.y, a2 = v0.z, a3 = v0.w;
  float a4 = v1.x, a5 = v1.y, a6 = v1.z, a7 = v1.w;
  float x0 = a0, x1 = a1;
  int id = 0;
  bool c;
  c = (a1 < x); id += c; x0 = c ? a1 : x0; x1 = c ? a2 : x1;
  c = (a2 < x); id += c; x0 = c ? a2 : x0; x1 = c ? a3 : x1;
  c = (a3 < x); id += c; x0 = c ? a3 : x0; x1 = c ? a4 : x1;
  c = (a4 < x); id += c; x0 = c ? a4 : x0; x1 = c ? a5 : x1;
  c = (a5 < x); id += c; x0 = c ? a5 : x0; x1 = c ? a6 : x1;
  c = (a6 < x); id += c; x0 = c ? a6 : x0; x1 = c ? a7 : x1;
  idx = id;
  tf = (x - x0) / (x1 - x0);   // extrapolates outside table range, as reference
}

// ---------- workspace init ----------
__global__ __launch_bounds__(256) void k_init(unsigned* __restrict__ maxkey,
                                              float* __restrict__ sums, int n) {
  int i = blockIdx.x * 256 + threadIdx.x;
  if (i < n) { maxkey[i] = 0u; sums[i] = 0.0f; }
}

// ---------- main timing evaluation: PHASE 0 = atomic max, PHASE 1 = atomic exp-sum ----------
template <int PHASE>
__global__ __launch_bounds__(256) void k_timing(
    const float* __restrict__ in_arrs,  const float* __restrict__ in_slews,
    const float* __restrict__ c1,       const float* __restrict__ c2,
    const int*   __restrict__ arc_r,    const int*   __restrict__ arc_f,
    const int*   __restrict__ group,    const int*   __restrict__ g_unate,
    const float* __restrict__ dtab,     const float* __restrict__ stab,
    const float* __restrict__ lidx,     const float* __restrict__ sidx,
    unsigned*    __restrict__ maxkey,   float* __restrict__ sums,
    int Tn, int An) {
  // Stage unateness[] (A=4096 ints, 16 KB) into LDS via CDNA5 async-to-LDS DMA.
  // Each lane issues 4x B128 (64B); INST_OFFSET advances both LDS and global addr.
  __shared__ int s_unate[4096];
  {
    unsigned lbase = (unsigned)(uintptr_t)(void*)s_unate;  // low 32b of generic ptr == LDS byte addr
    for (int e = (int)threadIdx.x * 16; e < An; e += 256 * 16) {
      unsigned lds = lbase + (unsigned)e * 4u;
      unsigned go  = (unsigned)e * 4u;
      asm volatile(
          "global_load_async_to_lds_b128 %0, %1, %2 offset:0\n\t"
          "global_load_async_to_lds_b128 %0, %1, %2 offset:16\n\t"
          "global_load_async_to_lds_b128 %0, %1, %2 offset:32\n\t"
          "global_load_async_to_lds_b128 %0, %1, %2 offset:48"
          :: "v"(lds), "v"(go), "s"(g_unate)
          : "memory");
    }
    asm volatile("s_wait_asynccnt 0" ::: "memory");
    __syncthreads();
  }

  int t = blockIdx.x * 256 + threadIdx.x;
  if (t >= Tn) return;

  float2 arr2 = ((const float2*)in_arrs)[t];   // [T,2] row, 8B aligned
  float2 slw2 = ((const float2*)in_slews)[t];
  float load = (c1[t] + c2[t]) / 1e15f;        // lumped load (use_ceff=False; rpi unused)
  int arc0 = arc_r[t], arc1 = arc_f[t];

  float v[4];
#pragma unroll
  for (int s = 0; s < 2; ++s) {
    int arc = (s == 0) ? arc0 : arc1;
    int un  = s_unate[arc] & 1;
    int rf  = un ^ s;                          // unate XOR invert(fall slot)
    float in_slew = rf ? slw2.y : slw2.x;
    float in_arr  = rf ? arr2.y : arr2.x;

    const float4* sp = (const float4*)(sidx + (size_t)arc * 8);  // 32B-aligned rows
    const float4* lp = (const float4*)(lidx + (size_t)arc * 8);
    float4 s0 = sp[0], s1 = sp[1];
    float4 l0 = lp[0], l1 = lp[1];

    int si, ci; float ts, tc;
    locate8(s0, s1, in_slew, si, ts);
    locate8(l0, l1, load,    ci, tc);

    const float* dp = dtab + (size_t)arc * 64 + si * 8 + ci;
    const float* zp = stab + (size_t)arc * 64 + si * 8 + ci;
    float w00 = (1.0f - ts) * (1.0f - tc);
    float w01 = (1.0f - ts) * tc;
    float w10 = ts * (1.0f - tc);
    float w11 = ts * tc;
    float d = w00 * dp[0] + w01 * dp[1] + w10 * dp[8] + w11 * dp[9];
    float z = w00 * zp[0] + w01 * zp[1] + w10 * zp[8] + w11 * zp[9];
    v[s]     = d + in_arr;
    v[2 + s] = z;
  }

  // nan_to_num(nan=-1e30, posinf=1e30, neginf=-1e30)
#pragma unroll
  for (int j = 0; j < 4; ++j) {
    float x = v[j];
    x = (x != x) ? -1e30f : x;
    x = fminf(fmaxf(x, -1e30f), 1e30f);
    v[j] = x;
  }

  int g = group[t];
  if (PHASE == 0) {
#pragma unroll
    for (int j = 0; j < 4; ++j)
      atomicMax(maxkey + (size_t)g * 4 + j, fkey(v[j]));   // global_atomic_max_u32 (exact)
  } else {
#pragma unroll
    for (int j = 0; j < 4; ++j) {
      unsigned k = maxkey[(size_t)g * 4 + j];
      float shift = (k == 0u) ? 0.0f : fkey_inv(k);        // key 0 == empty-group sentinel
      unsafeAtomicAdd(sums + (size_t)g * 4 + j,
                      __expf((v[j] - shift) * LSE_BETA));   // global_atomic_add_f32
    }
  }
}

// ---------- finalize: shift + log(max(sum,1e-30))/beta ----------
__global__ __launch_bounds__(256) void k_final(const unsigned* __restrict__ maxkey,
                                               const float* __restrict__ sums,
                                               float* __restrict__ out, int n) {
  int i = blockIdx.x * 256 + threadIdx.x;
  if (i < n) {
    unsigned k = maxkey[i];
    float shift = (k == 0u) ? 0.0f : fkey_inv(k);
    out[i] = shift + logf(fmaxf(sums[i], 1e-30f)) / LSE_BETA;
  }
}

extern "C" void kernel_launch(void* const* d_in, const int* in_sizes, int n_in,
                              void* d_out, int out_size, void* d_ws, size_t ws_size,
                              hipStream_t stream) {
  const float* in_arrs  = (const float*)d_in[0];
  const float* in_slews = (const float*)d_in[1];
  const float* c1       = (const float*)d_in[2];
  const float* c2       = (const float*)d_in[3];
  // d_in[4] = rpi (unused: use_ceff == False path)
  const int*   arc_r    = (const int*)d_in[5];
  const int*   arc_f    = (const int*)d_in[6];
  const int*   group    = (const int*)d_in[7];
  const int*   unate    = (const int*)d_in[8];
  const float* dtab     = (const float*)d_in[9];
  const float* stab     = (const float*)d_in[10];
  const float* lidx     = (const float*)d_in[11];  // load_index
  const float* sidx     = (const float*)d_in[12];  // slew_index

  const int Tn = in_sizes[2];     // c1 length == T
  const int An = in_sizes[8];     // unateness length == A (4096)
  const int G4 = out_size;        // G*4

  unsigned* maxkey = (unsigned*)d_ws;                                  // G4 u32
  float*    sums   = (float*)((char*)d_ws + (size_t)G4 * sizeof(unsigned)); // G4 f32
  float*    out    = (float*)d_out;

  int gridG = (G4 + 255) / 256;
  int gridT = (Tn + 255) / 256;

  k_init<<<gridG, 256, 0, stream>>>(maxkey, sums, G4);
  k_timing<0><<<gridT, 256, 0, stream>>>(in_arrs, in_slews, c1, c2, arc_r, arc_f,
                                         group, unate, dtab, stab, lidx, sidx,
                                         maxkey, sums, Tn, An);
  k_timing<1><<<gridT, 256, 0, stream>>>(in_arrs, in_slews, c1, c2, arc_r, arc_f,
                                         group, unate, dtab, stab, lidx, sidx,
                                         maxkey, sums, Tn, An);
  k_final<<<gridG, 256, 0, stream>>>(maxkey, sums, out, G4);
}